// GroupedVectorAttention_15229954031656
// MI455X (gfx1250) — compile-verified
//
#include <hip/hip_runtime.h>
#include <math.h>

#define CCH 256
#define KNB 16
#define GRP 16
#define EPS 1e-5f

typedef float v2f __attribute__((ext_vector_type(2)));
typedef float v8f __attribute__((ext_vector_type(8)));

static __device__ __forceinline__ v8f wmma4(v2f a, v2f b, v8f c) {
  // D = A(16x4) * B(4x16) + C(16x16), all f32, wave32
  return __builtin_amdgcn_wmma_f32_16x16x4_f32(false, a, false, b, (short)0, c, false, false);
}

// ---------------------------------------------------------------- utilities
__global__ void zero_kernel(float* p, int n) {
  int i = blockIdx.x * 256 + threadIdx.x;
  if (i < n) p[i] = 0.f;
}

// Wt[n][k] = W[k][n]
__global__ void transpose_kernel(const float* __restrict__ W, float* __restrict__ Wt,
                                 int Kd, int Nd) {
  int i = blockIdx.x * 256 + threadIdx.x;
  if (i < Kd * Nd) {
    int k = i / Nd, n = i % Nd;
    Wt[(size_t)n * Kd + k] = W[i];
  }
}

// ------------------------------------------------- generic WMMA f32 GEMM
// Y[M,256] = X[M,256] @ W[256,256] + bias, W given transposed (Wt[n][k]).
// grid=(M/16, 256/(32*GW)), block=32*GW.  X tile staged in LDS once per block.
#define GW 4
__global__ void gemm_kernel(const float* __restrict__ X, const float* __restrict__ Wt,
                            const float* __restrict__ bias, float* __restrict__ Y) {
  const int Kd = CCH, Nd = CCH;
  __shared__ float xl[16 * CCH];
  int tid = threadIdx.x;
  int lane = tid & 31, wave = tid >> 5;
  int m0 = blockIdx.x * 16;
  for (int e = tid * 4; e < 16 * Kd; e += (32 * GW) * 4) {
    int row = e >> 8, col = e & 255;
    *(float4*)&xl[e] = *(const float4*)&X[(size_t)(m0 + row) * Kd + col];
  }
  __syncthreads();
  int n0 = (blockIdx.y * GW + wave) * 32;
  int kh = (lane >> 4) * 2;                 // A/B K-half per ISA layout
  const float* xr = &xl[(lane & 15) * Kd];
  int col0 = n0 + (lane & 15), col1 = col0 + 16;
  const float* wt0 = Wt + (size_t)col0 * Kd;
  const float* wt1 = Wt + (size_t)col1 * Kd;
  v8f c0 = {}; v8f c1 = {};
  for (int kk = 0; kk < Kd; kk += 4) {
    v2f a  = *(const v2f*)&xr[kk + kh];      // ds_load_b64
    v2f b0 = *(const v2f*)&wt0[kk + kh];     // global_load_b64
    v2f b1 = *(const v2f*)&wt1[kk + kh];
    c0 = wmma4(a, b0, c0);
    c1 = wmma4(a, b1, c1);
  }
  int cm = (lane >> 4) * 8, cn = lane & 15;
  for (int v = 0; v < 8; ++v) {
    size_t r = (size_t)(m0 + cm + v) * Nd;
    Y[r + n0 + cn]      = c0[v] + bias[n0 + cn];
    Y[r + n0 + 16 + cn] = c1[v] + bias[n0 + 16 + cn];
  }
}

// gate_in = concat(q, nctx, gctx[seg]); Kd=768 staged in LDS; cg_w1 transposed.
__global__ void gemm_gate_kernel(const float* __restrict__ q, const float* __restrict__ nctx,
                                 const float* __restrict__ gctx, const int* __restrict__ offset,
                                 const float* __restrict__ Wt, const float* __restrict__ bias,
                                 float* __restrict__ Y) {
  const int Kd = 3 * CCH, Nd = CCH;
  __shared__ float xl[16 * 3 * CCH];
  int tid = threadIdx.x;
  int lane = tid & 31, wave = tid >> 5;
  int m0 = blockIdx.x * 16;
  int o0 = offset[0], o1 = offset[1], o2 = offset[2];
  for (int e = tid; e < 16 * Kd; e += 32 * GW) {
    int row = m0 + e / Kd;
    int col = e % Kd;
    float val;
    if (col < CCH)          val = q[(size_t)row * CCH + col];
    else if (col < 2 * CCH) val = nctx[(size_t)row * CCH + col - CCH];
    else {
      int seg = (row >= o0) + (row >= o1) + (row >= o2);
      val = gctx[seg * CCH + col - 2 * CCH];
    }
    xl[e] = val;
  }
  __syncthreads();
  int n0 = (blockIdx.y * GW + wave) * 32;
  int kh = (lane >> 4) * 2;
  const float* xr = &xl[(lane & 15) * Kd];
  int col0 = n0 + (lane & 15), col1 = col0 + 16;
  const float* wt0 = Wt + (size_t)col0 * Kd;
  const float* wt1 = Wt + (size_t)col1 * Kd;
  v8f c0 = {}; v8f c1 = {};
  for (int kk = 0; kk < Kd; kk += 4) {
    v2f a  = *(const v2f*)&xr[kk + kh];
    v2f b0 = *(const v2f*)&wt0[kk + kh];
    v2f b1 = *(const v2f*)&wt1[kk + kh];
    c0 = wmma4(a, b0, c0);
    c1 = wmma4(a, b1, c1);
  }
  int cm = (lane >> 4) * 8, cn = lane & 15;
  for (int v = 0; v < 8; ++v) {
    size_t r = (size_t)(m0 + cm + v) * Nd;
    Y[r + n0 + cn]      = c0[v] + bias[n0 + cn];
    Y[r + n0 + 16 + cn] = c1[v] + bias[n0 + 16 + cn];
  }
}

// ------------------------------------------------- per-channel BN statistics
__global__ void colstats_kernel(const float* __restrict__ X, int rows, float* __restrict__ st) {
  int c = threadIdx.x;  // 256 threads = channels
  float s = 0.f, s2 = 0.f;
  for (int r = blockIdx.x; r < rows; r += gridDim.x) {
    float x = X[(size_t)r * CCH + c];
    s += x; s2 += x * x;
  }
  atomicAdd(&st[c], s);
  atomicAdd(&st[CCH + c], s2);
}

// BN+ReLU in place for q and k
__global__ void bnrelu2_kernel(float* q, float* k, const float* stq, const float* stk,
                               const float* qg, const float* qbeta,
                               const float* kg, const float* kbeta, float invN) {
  int row = blockIdx.x, c = threadIdx.x;
  size_t i = (size_t)row * CCH + c;
  float mq = stq[c] * invN, vq = stq[CCH + c] * invN - mq * mq;
  float x = q[i];
  q[i] = fmaxf((x - mq) * rsqrtf(vq + EPS) * qg[c] + qbeta[c], 0.f);
  float mk = stk[c] * invN, vk = stk[CCH + c] * invN - mk * mk;
  float y = k[i];
  k[i] = fmaxf((y - mk) * rsqrtf(vk + EPS) * kg[c] + kbeta[c], 0.f);
}

// neighbor_ctx = max_s q[idx]*mf
__global__ void nctx_kernel(const float* __restrict__ q, const int* __restrict__ refidx,
                            float* __restrict__ nctx) {
  int n = blockIdx.x, c = threadIdx.x;
  float m = 0.f;  // q >= 0 post-ReLU so 0 is identity wrt masked zeros
  for (int s = 0; s < KNB; ++s) {
    int idx = refidx[n * KNB + s];
    float v = (idx >= 0) ? q[(size_t)idx * CCH + c] : 0.f;
    m = fmaxf(m, v);
  }
  nctx[(size_t)n * CCH + c] = m;
}

// per-segment max of q (4 segments).  q>=0 so uint-bit atomicMax is valid.
__global__ void gctx_kernel(const float* __restrict__ q, const int* __restrict__ offset,
                            float* __restrict__ gctx, int N) {
  int c = threadIdx.x;
  int chunk = (N + gridDim.x - 1) / gridDim.x;
  int r0 = blockIdx.x * chunk;
  int r1 = min(r0 + chunk, N);
  int o0 = offset[0], o1 = offset[1], o2 = offset[2];
  float m = 0.f; int cur = -1;
  for (int n = r0; n < r1; ++n) {
    int seg = (n >= o0) + (n >= o1) + (n >= o2);
    if (seg != cur) {
      if (cur >= 0) atomicMax((unsigned int*)&gctx[cur * CCH + c], __float_as_uint(m));
      cur = seg; m = 0.f;
    }
    m = fmaxf(m, q[(size_t)n * CCH + c]);
  }
  if (cur >= 0) atomicMax((unsigned int*)&gctx[cur * CCH + c], __float_as_uint(m));
}

// gate = sigmoid(relu(bn(h1)) @ cg_w2 + cg_b2)  -- one wave per 16 points, WMMA
__global__ void gate_kernel(const float* __restrict__ h1, const float* __restrict__ st,
                            const float* __restrict__ gma, const float* __restrict__ bta,
                            const float* __restrict__ cg_w2, const float* __restrict__ cg_b2,
                            float* __restrict__ gate, float invN) {
  int lane = threadIdx.x;
  int p0 = blockIdx.x * 16;
  int row = p0 + (lane & 15);
  int kh = (lane >> 4) * 2;
  int gcol = lane & 15;
  v8f c = {};
  for (int kk = 0; kk < CCH; kk += 4) {
    int c0 = kk + kh, c1 = c0 + 1;
    float m0 = st[c0] * invN, v0 = st[CCH + c0] * invN - m0 * m0;
    float m1 = st[c1] * invN, v1 = st[CCH + c1] * invN - m1 * m1;
    v2f a, b;
    a.x = fmaxf((h1[(size_t)row * CCH + c0] - m0) * rsqrtf(v0 + EPS) * gma[c0] + bta[c0], 0.f);
    a.y = fmaxf((h1[(size_t)row * CCH + c1] - m1) * rsqrtf(v1 + EPS) * gma[c1] + bta[c1], 0.f);
    b.x = cg_w2[c0 * GRP + gcol];
    b.y = cg_w2[c1 * GRP + gcol];
    c = wmma4(a, b, c);
  }
  int cm = (lane >> 4) * 8;
  for (int v = 0; v < 8; ++v) {
    float x = c[v] + cg_b2[gcol];
    gate[(size_t)(p0 + cm + v) * GRP + gcol] = 1.f / (1.f + __expf(-x));
  }
}

// Wf = pb_w2 @ we_w1 [256,16] ; cb = pb_b2 @ we_w1 + we_b1 [16]
__global__ void wf_kernel(const float* __restrict__ pb_w2, const float* __restrict__ we_w1,
                          const float* __restrict__ we_b1, const float* __restrict__ pb_b2,
                          float* __restrict__ Wf, float* __restrict__ cb) {
  int j = threadIdx.x;
  for (int g = 0; g < GRP; ++g) {
    float s = 0.f;
    for (int c = 0; c < CCH; ++c) s += pb_w2[(size_t)j * CCH + c] * we_w1[c * GRP + g];
    Wf[j * GRP + g] = s;
  }
  if (j < GRP) {
    float s = we_b1[j];
    for (int c = 0; c < CCH; ++c) s += pb_b2[c] * we_w1[c * GRP + j];
    cb[j] = s;
  }
}

// stats of ph = pos @ pb_w1 + pb_b1 over all N*K rows (recompute, never materialize)
__global__ void pbstats_kernel(const float* __restrict__ coord, const int* __restrict__ refidx,
                               const float* __restrict__ pb_w1, const float* __restrict__ pb_b1,
                               float* __restrict__ st, int NK) {
  int c = threadIdx.x;
  float w0 = pb_w1[c], w1 = pb_w1[CCH + c], w2 = pb_w1[2 * CCH + c], b = pb_b1[c];
  float s = 0.f, s2 = 0.f;
  for (int r = blockIdx.x; r < NK; r += gridDim.x) {
    int n = r >> 4, ss = r & 15;
    int idx = refidx[n * KNB + ss];
    float px = 0.f, py = 0.f, pz = 0.f;
    if (idx >= 0) {
      px = coord[idx * 3 + 0] - coord[n * 3 + 0];
      py = coord[idx * 3 + 1] - coord[n * 3 + 1];
      pz = coord[idx * 3 + 2] - coord[n * 3 + 2];
    }
    float ph = px * w0 + py * w1 + pz * w2 + b;
    s += ph; s2 += ph * ph;
  }
  atomicAdd(&st[c], s);
  atomicAdd(&st[CCH + c], s2);
}

// ---------------------------- per-point helpers (one wave per point) --------
__device__ __forceinline__ void build_pos_lds(float* pos, const float* coord,
                                              const int* refidx, int n, int lane) {
  if (lane < 16) {
    int idx = refidx[n * KNB + lane];
    float px = 0.f, py = 0.f, pz = 0.f;
    if (idx >= 0) {
      px = coord[idx * 3 + 0] - coord[n * 3 + 0];
      py = coord[idx * 3 + 1] - coord[n * 3 + 1];
      pz = coord[idx * 3 + 2] - coord[n * 3 + 2];
    }
    pos[lane * 4 + 0] = px; pos[lane * 4 + 1] = py; pos[lane * 4 + 2] = pz;
  }
}

// prefetch a gathered 1KB row (4 cachelines assumed 256B)
__device__ __forceinline__ void prefetch_row(const float* row) {
  __builtin_prefetch(row, 0, 0);
  __builtin_prefetch(row + 64, 0, 0);
  __builtin_prefetch(row + 128, 0, 0);
  __builtin_prefetch(row + 192, 0, 0);
}

// r[s][c] = relu(bn(pos@pb_w1 + pb_b1))  (16 x 256 tile in LDS)
__device__ __forceinline__ void build_r_lds(float* r, const float* pos,
                                            const float* pb_w1, const float* pb_b1,
                                            const float* st_pb, const float* pb_g,
                                            const float* pb_beta, float invNK, int lane) {
  for (int e = lane; e < 16 * CCH; e += 32) {
    int s = e >> 8, c = e & 255;
    float ph = pos[s * 4 + 0] * pb_w1[c] + pos[s * 4 + 1] * pb_w1[CCH + c] +
               pos[s * 4 + 2] * pb_w1[2 * CCH + c] + pb_b1[c];
    float m = st_pb[c] * invNK;
    float v = st_pb[CCH + c] * invNK - m * m;
    float y = (ph - m) * rsqrtf(v + EPS) * pb_g[c] + pb_beta[c];
    r[e] = fmaxf(y, 0.f);
  }
}

// hwe[16s,16g] = (k[idx]*mf - q) @ we_w1 + r @ Wf   (cb[g] added by caller)
__device__ __forceinline__ v8f hwe_wmma(const float* kbuf, const float* qbuf, const float* r,
                                        const int* refidx, int n,
                                        const float* we_w1, const float* Wf, int lane) {
  int srow = lane & 15;
  int kh = (lane >> 4) * 2;
  int gcol = lane & 15;
  int idx = refidx[n * KNB + srow];
  bool valid = idx >= 0;
  const float* krow = kbuf + (size_t)(valid ? idx : 0) * CCH;
  const float* qrow = qbuf + (size_t)n * CCH;
  v8f c = {};
  for (int kk = 0; kk < CCH; kk += 4) {
    int c0 = kk + kh, c1 = c0 + 1;
    v2f a1, a2, b1, b2;
    a1.x = (valid ? krow[c0] : 0.f) - qrow[c0];
    a1.y = (valid ? krow[c1] : 0.f) - qrow[c1];
    a2.x = r[srow * CCH + c0];
    a2.y = r[srow * CCH + c1];
    b1.x = we_w1[c0 * GRP + gcol]; b1.y = we_w1[c1 * GRP + gcol];
    b2.x = Wf[c0 * GRP + gcol];    b2.y = Wf[c1 * GRP + gcol];
    c = wmma4(a1, b1, c);
    c = wmma4(a2, b2, c);
  }
  return c;
}

// stats of hwe over all (n,s) rows, 16 channels
__global__ void westats_kernel(const float* __restrict__ coord, const int* __restrict__ refidx,
                               const float* __restrict__ kbuf, const float* __restrict__ qbuf,
                               const float* __restrict__ pb_w1, const float* __restrict__ pb_b1,
                               const float* __restrict__ st_pb, const float* __restrict__ pb_g,
                               const float* __restrict__ pb_beta,
                               const float* __restrict__ we_w1, const float* __restrict__ Wf,
                               const float* __restrict__ cb, float* __restrict__ st_we,
                               float invNK) {
  __shared__ float r[16 * CCH];
  __shared__ float pos[16 * 4];
  int lane = threadIdx.x;
  int n = blockIdx.x;
  {  // warm the gathered k rows while we build the r tile
    int idx = refidx[n * KNB + (lane & 15)];
    if (idx >= 0) prefetch_row(kbuf + (size_t)idx * CCH);
  }
  build_pos_lds(pos, coord, refidx, n, lane);
  __syncthreads();
  build_r_lds(r, pos, pb_w1, pb_b1, st_pb, pb_g, pb_beta, invNK, lane);
  __syncthreads();
  v8f h = hwe_wmma(kbuf, qbuf, r, refidx, n, we_w1, Wf, lane);
  int g = lane & 15;
  float cbg = cb[g];
  float s = 0.f, s2 = 0.f;
  for (int v = 0; v < 8; ++v) {
    float x = h[v] + cbg;
    s += x; s2 += x * x;
  }
  s  += __shfl_xor(s, 16);
  s2 += __shfl_xor(s2, 16);
  if (lane < 16) {
    atomicAdd(&st_we[lane], s);
    atomicAdd(&st_we[GRP + lane], s2);
  }
}

// -------------------------------- final fused kernel: one wave per point ----
__global__ void final_kernel(const float* __restrict__ coord, const int* __restrict__ refidx,
                             const float* __restrict__ kbuf, const float* __restrict__ qbuf,
                             const float* __restrict__ vbuf, const float* __restrict__ gate,
                             const float* __restrict__ pb_w1, const float* __restrict__ pb_b1,
                             const float* __restrict__ st_pb, const float* __restrict__ pb_g,
                             const float* __restrict__ pb_beta,
                             const float* __restrict__ pb_w2, const float* __restrict__ pb_b2,
                             const float* __restrict__ we_w1, const float* __restrict__ Wf,
                             const float* __restrict__ cb, const float* __restrict__ st_we,
                             const float* __restrict__ we_g, const float* __restrict__ we_beta,
                             const float* __restrict__ we_w2, const float* __restrict__ we_b2,
                             float* __restrict__ out, float invNK) {
  __shared__ float r[16 * CCH];   // r tile, later overwritten by t[g][j]
  __shared__ float pos[16 * 4];
  __shared__ float rh[16 * GRP];  // relu(bn(hwe)) [s][g]
  __shared__ float wt[16 * GRP];  // softmax weights [s][g]
  __shared__ float sw[GRP];
  int lane = threadIdx.x;
  int n = blockIdx.x;

  {  // warm gathered k rows (lanes 0-15) and v rows (lanes 16-31)
    int idx = refidx[n * KNB + (lane & 15)];
    if (idx >= 0) {
      const float* row = (lane < 16) ? (kbuf + (size_t)idx * CCH) : (vbuf + (size_t)idx * CCH);
      prefetch_row(row);
    }
  }
  build_pos_lds(pos, coord, refidx, n, lane);
  __syncthreads();
  build_r_lds(r, pos, pb_w1, pb_b1, st_pb, pb_g, pb_beta, invNK, lane);
  __syncthreads();

  // 1. weight logits hwe -> BN -> relu -> rh
  v8f h = hwe_wmma(kbuf, qbuf, r, refidx, n, we_w1, Wf, lane);
  int g = lane & 15;
  int sbase = (lane >> 4) * 8;
  {
    float m = st_we[g] * invNK;
    float va = st_we[GRP + g] * invNK - m * m;
    float inv = rsqrtf(va + EPS);
    float cbg = cb[g];
    for (int v = 0; v < 8; ++v) {
      float x = h[v] + cbg;
      rh[(sbase + v) * GRP + g] = fmaxf((x - m) * inv * we_g[g] + we_beta[g], 0.f);
    }
  }
  __syncthreads();

  // 2. weight = rh @ we_w2 + we_b2, * gate, softmax over s, mask
  float w[8];
  float gv = gate[(size_t)n * GRP + g];
  for (int v = 0; v < 8; ++v) {
    float acc = we_b2[g];
    for (int gp = 0; gp < GRP; ++gp) acc += rh[(sbase + v) * GRP + gp] * we_w2[gp * GRP + g];
    w[v] = acc * gv;
  }
  float mx = w[0];
  for (int v = 1; v < 8; ++v) mx = fmaxf(mx, w[v]);
  mx = fmaxf(mx, __shfl_xor(mx, 16));
  float sum = 0.f;
  for (int v = 0; v < 8; ++v) { w[v] = __expf(w[v] - mx); sum += w[v]; }
  sum += __shfl_xor(sum, 16);
  float rs = 1.f / sum;
  for (int v = 0; v < 8; ++v) {
    int s = sbase + v;
    int idx = refidx[n * KNB + s];
    wt[s * GRP + g] = (idx >= 0) ? w[v] * rs : 0.f;
  }
  __syncthreads();
  if (lane < 16) {
    float acc = 0.f;
    for (int s = 0; s < 16; ++s) acc += wt[s * GRP + lane];
    sw[lane] = acc;
  }
  __syncthreads();

  // 3. t[g][j] = sum_s wt[s][g] * r[s][j]  via WMMA (M=g, N=j-tile, K=s); overwrite r
  {
    int kh = (lane >> 4) * 2;
    int lcol = lane & 15;
    for (int jt = 0; jt < 16; ++jt) {
      int j0 = jt * 16;
      v8f t = {};
      for (int kk = 0; kk < 16; kk += 4) {
        int s0 = kk + kh, s1 = s0 + 1;
        v2f a, b;
        a.x = wt[s0 * GRP + lcol]; a.y = wt[s1 * GRP + lcol];   // A[g=lcol, s]
        b.x = r[s0 * CCH + j0 + lcol]; b.y = r[s1 * CCH + j0 + lcol];
        t = wmma4(a, b, t);
      }
      int gm = (lane >> 4) * 8;
      for (int v = 0; v < 8; ++v) r[(gm + v) * CCH + j0 + lcol] = t[v];
    }
  }
  __syncthreads();

  // 4. out[c] = sum_s wt[s,g]*v[idx_s,c] + sum_j t[g,j]*pb_w2[j,c] + pb_b2[c]*sw[g]
  int cbase = lane * 8;             // 8 contiguous channels, same group per lane
  int gl = lane >> 1;               // group of this lane's channels
  float acc[8];
  for (int u = 0; u < 8; ++u) acc[u] = pb_b2[cbase + u] * sw[gl];
  for (int s = 0; s < 16; ++s) {
    int idx = refidx[n * KNB + s];
    if (idx < 0) continue;
    float wsg = wt[s * GRP + gl];
    const float* vrow = vbuf + (size_t)idx * CCH + cbase;
    for (int u = 0; u < 8; ++u) acc[u] += wsg * vrow[u];
  }
  const float* trow = &r[gl * CCH];
  for (int j = 0; j < CCH; ++j) {
    float tv = trow[j];
    const float* wr = pb_w2 + (size_t)j * CCH + cbase;
    for (int u = 0; u < 8; ++u) acc[u] += tv * wr[u];
  }
  float* orow = out + (size_t)n * CCH + cbase;
  for (int u = 0; u < 8; ++u) orow[u] = acc[u];
}

// ---------------------------------------------------------------------------
extern "C" void kernel_launch(void* const* d_in, const int* in_sizes, int n_in,
                              void* d_out, int out_size, void* d_ws, size_t ws_size,
                              hipStream_t stream) {
  const float* feat    = (const float*)d_in[0];
  const float* coord   = (const float*)d_in[1];
  const int*   offset  = (const int*)d_in[2];
  const int*   refidx  = (const int*)d_in[3];
  const float* wq = (const float*)d_in[4],  * bq = (const float*)d_in[5];
  const float* qg = (const float*)d_in[6],  * qbeta = (const float*)d_in[7];
  const float* wk = (const float*)d_in[8],  * bk = (const float*)d_in[9];
  const float* kg = (const float*)d_in[10], * kbeta = (const float*)d_in[11];
  const float* wv = (const float*)d_in[12], * bv = (const float*)d_in[13];
  const float* pb_w1 = (const float*)d_in[14], * pb_b1 = (const float*)d_in[15];
  const float* pb_g  = (const float*)d_in[16], * pb_beta = (const float*)d_in[17];
  const float* pb_w2 = (const float*)d_in[18], * pb_b2 = (const float*)d_in[19];
  const float* we_w1 = (const float*)d_in[20], * we_b1 = (const float*)d_in[21];
  const float* we_g  = (const float*)d_in[22], * we_beta = (const float*)d_in[23];
  const float* we_w2 = (const float*)d_in[24], * we_b2 = (const float*)d_in[25];
  const float* cg_w1 = (const float*)d_in[26], * cg_b1 = (const float*)d_in[27];
  const float* cg_g  = (const float*)d_in[28], * cg_beta = (const float*)d_in[29];
  const float* cg_w2 = (const float*)d_in[30], * cg_b2 = (const float*)d_in[31];
  float* out = (float*)d_out;

  const int N = in_sizes[0] / CCH;
  const int NK = N * KNB;
  const float invN  = 1.0f / (float)N;
  const float invNK = 1.0f / (float)NK;

  // workspace layout
  float* W = (float*)d_ws;
  size_t NC = (size_t)N * CCH;
  float* qbuf = W;
  float* kbuf = qbuf + NC;
  float* vbuf = kbuf + NC;
  float* nctx = vbuf + NC;
  float* h1   = nctx + NC;
  float* gate = h1 + NC;                 // N*16
  float* gctx = gate + (size_t)N * GRP;  // 4*256
  float* st_q  = gctx + 4 * CCH;         // 512
  float* st_k  = st_q + 2 * CCH;
  float* st_h1 = st_k + 2 * CCH;
  float* st_pb = st_h1 + 2 * CCH;
  float* st_we = st_pb + 2 * CCH;        // 32
  float* Wf    = st_we + 2 * GRP;        // 256*16
  float* cb    = Wf + CCH * GRP;         // 16
  float* wqt   = cb + GRP;               // 256*256 transposed weights
  float* wkt   = wqt + CCH * CCH;
  float* wvt   = wkt + CCH * CCH;
  float* cgw1t = wvt + CCH * CCH;        // 768*256

  // 0. zero gctx + all stats (contiguous: gctx .. st_we end)
  int zcount = 4 * CCH + 4 * (2 * CCH) + 2 * GRP;
  zero_kernel<<<(zcount + 255) / 256, 256, 0, stream>>>(gctx, zcount);

  // 0b. transpose weights for b64 B-operand loads
  transpose_kernel<<<(CCH * CCH + 255) / 256, 256, 0, stream>>>(wq, wqt, CCH, CCH);
  transpose_kernel<<<(CCH * CCH + 255) / 256, 256, 0, stream>>>(wk, wkt, CCH, CCH);
  transpose_kernel<<<(CCH * CCH + 255) / 256, 256, 0, stream>>>(wv, wvt, CCH, CCH);
  transpose_kernel<<<(3 * CCH * CCH + 255) / 256, 256, 0, stream>>>(cg_w1, cgw1t, 3 * CCH, CCH);

  // 1. q/k/v linear projections (WMMA f32 GEMM, LDS-staged A)
  dim3 ggrid(N / 16, CCH / (32 * GW));
  gemm_kernel<<<ggrid, 32 * GW, 0, stream>>>(feat, wqt, bq, qbuf);
  gemm_kernel<<<ggrid, 32 * GW, 0, stream>>>(feat, wkt, bk, kbuf);
  gemm_kernel<<<ggrid, 32 * GW, 0, stream>>>(feat, wvt, bv, vbuf);

  // 2. BN stats for q,k; apply BN+ReLU
  colstats_kernel<<<2048, CCH, 0, stream>>>(qbuf, N, st_q);
  colstats_kernel<<<2048, CCH, 0, stream>>>(kbuf, N, st_k);
  bnrelu2_kernel<<<N, CCH, 0, stream>>>(qbuf, kbuf, st_q, st_k, qg, qbeta, kg, kbeta, invN);

  // 3. contexts + gating
  nctx_kernel<<<N, CCH, 0, stream>>>(qbuf, refidx, nctx);
  gctx_kernel<<<1024, CCH, 0, stream>>>(qbuf, offset, gctx, N);
  gemm_gate_kernel<<<ggrid, 32 * GW, 0, stream>>>(qbuf, nctx, gctx, offset, cgw1t, cg_b1, h1);
  colstats_kernel<<<2048, CCH, 0, stream>>>(h1, N, st_h1);
  gate_kernel<<<N / 16, 32, 0, stream>>>(h1, st_h1, cg_g, cg_beta, cg_w2, cg_b2, gate, invN);

  // 4. fused weights + pb BN stats + we BN stats
  wf_kernel<<<1, CCH, 0, stream>>>(pb_w2, we_w1, we_b1, pb_b2, Wf, cb);
  pbstats_kernel<<<4096, CCH, 0, stream>>>(coord, refidx, pb_w1, pb_b1, st_pb, NK);
  westats_kernel<<<N, 32, 0, stream>>>(coord, refidx, kbuf, qbuf, pb_w1, pb_b1, st_pb,
                                       pb_g, pb_beta, we_w1, Wf, cb, st_we, invNK);

  // 5. fused attention output, one wave per point
  final_kernel<<<N, 32, 0, stream>>>(coord, refidx, kbuf, qbuf, vbuf, gate,
                                     pb_w1, pb_b1, st_pb, pb_g, pb_beta, pb_w2, pb_b2,
                                     we_w1, Wf, cb, st_we, we_g, we_beta, we_w2, we_b2,
                                     out, invNK);
}